// PairwiseDotProduct_47725676593222
// MI455X (gfx1250) — compile-verified
//
#include <hip/hip_runtime.h>

// ---------------------------------------------------------------------------
// Row-wise dot product: out[i] = sum_k x1[i,k] * x2[i,k]
// B = 131072 rows, D = 256 cols, fp32. Pure HBM-bandwidth bound (~11.5us floor
// at 23.3 TB/s). Strategy: async global->LDS double-buffered streaming
// (ASYNCcnt pipeline), padded-LDS b128 reads, per-lane FMA partials, and a
// single v_wmma_f32_16x16x4_f32 per 16-row tile to fold the lane-pair
// partials (A-fragment K0/K2 slots) into the 16 row sums.
// ---------------------------------------------------------------------------

typedef float v2f __attribute__((ext_vector_type(2)));
typedef float v8f __attribute__((ext_vector_type(8)));
typedef int   v4i __attribute__((ext_vector_type(4)));

typedef __attribute__((address_space(1))) v4i gv4i;  // global (clang prints "__device__")
typedef __attribute__((address_space(3))) v4i lv4i;  // LDS    (clang prints "__shared__")

#define LDS_STRIDE 132   // 128 data floats + 4 pad -> conflict-free ds_load_b128

#if defined(__has_builtin)
#if __has_builtin(__builtin_amdgcn_global_load_async_to_lds_b128)
#define HAVE_ASYNC_LDS 1
#endif
#if __has_builtin(__builtin_amdgcn_s_wait_asynccnt)
#define HAVE_WAIT_ASYNC 1
#endif
#endif

__device__ __forceinline__ void async_cp16(const float* g, float* l) {
#if defined(HAVE_ASYNC_LDS)
  __builtin_amdgcn_global_load_async_to_lds_b128(
      (gv4i*)(v4i*)(float*)g,   // generic -> AS1, pointee v4i
      (lv4i*)(v4i*)l,           // generic -> AS3, pointee v4i
      /*offset=*/0, /*cpol=*/0);
#else
  *(float4*)l = *(const float4*)g;
#endif
}

__device__ __forceinline__ void wait_async_le32() {
#if defined(HAVE_WAIT_ASYNC)
  __builtin_amdgcn_s_wait_asynccnt(32);
#else
  asm volatile("s_wait_asynccnt 32" ::: "memory");
#endif
}

__device__ __forceinline__ void wait_async_le0() {
#if defined(HAVE_WAIT_ASYNC)
  __builtin_amdgcn_s_wait_asynccnt(0);
#else
  asm volatile("s_wait_asynccnt 0" ::: "memory");
#endif
}

// Issue one half-tile stage: 16 rows x 128 cols of x1 and x2 into LDS.
// 32 async b128 ops per wave; global addresses fully coalesced (512B/op).
__device__ __forceinline__ void stage_issue(const float* __restrict__ x1,
                                            const float* __restrict__ x2,
                                            float* l1, float* l2,
                                            int t, int s, int lane) {
  const size_t base = (size_t)t * (16 * 256) + (size_t)s * 128 + (size_t)lane * 4;
  const float* g1 = x1 + base;
  const float* g2 = x2 + base;
  float* d1 = l1 + lane * 4;
  float* d2 = l2 + lane * 4;
#pragma unroll
  for (int i = 0; i < 16; ++i) {
    async_cp16(g1 + i * 256, d1 + i * LDS_STRIDE);
    async_cp16(g2 + i * 256, d2 + i * LDS_STRIDE);
  }
}

// Consume one staged half-tile: lane l handles row (l&15); lane-halves take
// alternating float4 chunks so both halves cover every staged column block.
// 4 independent FMA chains to hide VALU latency.
__device__ __forceinline__ float stage_dot(const float* l1, const float* l2,
                                           int r, int h) {
  const float4* A = (const float4*)(l1 + r * LDS_STRIDE);
  const float4* Bv = (const float4*)(l2 + r * LDS_STRIDE);
  float s0 = 0.f, s1 = 0.f, s2 = 0.f, s3 = 0.f;
#pragma unroll
  for (int j = 0; j < 16; ++j) {
    const int idx = (j << 1) | h;   // chunk parity = lane half
    float4 a = A[idx];
    float4 b = Bv[idx];
    s0 = fmaf(a.x, b.x, s0);
    s1 = fmaf(a.y, b.y, s1);
    s2 = fmaf(a.z, b.z, s2);
    s3 = fmaf(a.w, b.w, s3);
  }
  return (s0 + s1) + (s2 + s3);
}

__global__ void __launch_bounds__(32)
rowdot_kernel(const float* __restrict__ x1, const float* __restrict__ x2,
              float* __restrict__ out, int nt) {
  // 2 buffers x 2 matrices x (16 rows * 132 floats) = 33,792 bytes LDS.
  __shared__ __align__(16) float lds[2][2][16 * LDS_STRIDE];

  const int lane = threadIdx.x;          // wave32 block
  const int r = lane & 15;               // row within 16-row tile
  const int h = lane >> 4;               // column-parity half
  const int stride = gridDim.x;

  int t = blockIdx.x;
  if (t >= nt) return;

  int cur = 0;
  stage_issue(x1, x2, &lds[cur][0][0], &lds[cur][1][0], t, 0, lane);

  float partial = 0.f;
  for (;;) {
    // ---- stage (t, 0) in buf `cur`; prefetch (t, 1) into the other buf ----
    stage_issue(x1, x2, &lds[cur ^ 1][0][0], &lds[cur ^ 1][1][0], t, 1, lane);
    wait_async_le32();                   // stage (t,0) complete
    partial += stage_dot(&lds[cur][0][0], &lds[cur][1][0], r, h);
    cur ^= 1;

    // ---- stage (t, 1); prefetch next tile's first half if any ----
    const int tn = t + stride;
    const bool more = tn < nt;
    if (more) {
      stage_issue(x1, x2, &lds[cur ^ 1][0][0], &lds[cur ^ 1][1][0], tn, 0, lane);
      wait_async_le32();                 // stage (t,1) complete
    } else {
      wait_async_le0();
    }
    partial += stage_dot(&lds[cur][0][0], &lds[cur][1][0], r, h);
    cur ^= 1;

    // ---- fold 16 rows with one f32 WMMA: row m = A[m,K0] + A[m,K2] -------
    // Lane m (m<16) carries K0, lane m+16 carries K2 in A VGPR0; VGPR1 = 0.
    // B = ones => D[m,n] = partial_h0(m) + partial_h1(m), replicated over N.
    v2f av; av[0] = partial; av[1] = 0.0f;
    v2f bv; bv[0] = 1.0f;    bv[1] = 1.0f;
    v8f cv = {};
    cv = __builtin_amdgcn_wmma_f32_16x16x4_f32(
        /*neg_a=*/false, av, /*neg_b=*/false, bv,
        /*c_mod=*/(short)0, cv, /*reuse_a=*/false, /*reuse_b=*/false);

    // D layout: VGPR k, lanes 0-15 -> M=k; lanes 16-31 -> M=k+8.
    if ((lane & 15) == 0) {
      float* o = out + (size_t)t * 16 + (h << 3);
      float4 lo4; lo4.x = cv[0]; lo4.y = cv[1]; lo4.z = cv[2]; lo4.w = cv[3];
      float4 hi4; hi4.x = cv[4]; hi4.y = cv[5]; hi4.z = cv[6]; hi4.w = cv[7];
      ((float4*)o)[0] = lo4;
      ((float4*)o)[1] = hi4;
    }

    partial = 0.f;
    if (!more) break;
    t = tn;
  }
}

extern "C" void kernel_launch(void* const* d_in, const int* in_sizes, int n_in,
                              void* d_out, int out_size, void* d_ws, size_t ws_size,
                              hipStream_t stream) {
  (void)n_in; (void)out_size; (void)d_ws; (void)ws_size;
  const float* x1 = (const float*)d_in[0];
  const float* x2 = (const float*)d_in[1];
  float* out = (float*)d_out;

  const int Brows = in_sizes[0] / 256;   // 131072
  const int nt = Brows / 16;             // 8192 16-row tiles
  const int grid = nt < 4096 ? nt : 4096;

  rowdot_kernel<<<grid, 32, 0, stream>>>(x1, x2, out, nt);
}